// RandomWalkGAT_12386685682215
// MI455X (gfx1250) — compile-verified
//
#include <hip/hip_runtime.h>
#include <math.h>

typedef __attribute__((ext_vector_type(2))) float v2f;
typedef __attribute__((ext_vector_type(8))) float v8f;

#define FIN  256
#define FOUT 256   // heads*hid for layer1 (4*64) and layer2 (1*256); also GEMM N
#define GK   256   // K dim of both GEMMs
#define APITCH 264 // LDS row pitch (floats): 16B-aligned float4 stores, spreads banks

// ---------------------------------------------------------------------------
// float atomic max via signed-max / unsigned-min bit trick (init must be -inf)
// ---------------------------------------------------------------------------
__device__ __forceinline__ void atomicMaxF(float* addr, float v) {
    if (v >= 0.0f) atomicMax((int*)addr, __float_as_int(v));
    else           atomicMin((unsigned int*)addr, __float_as_uint(v));
}

// ---------------------------------------------------------------------------
// fp32 WMMA GEMM: C[M,256] = A[M,256] @ B[256,256], row-major, M%16==0.
// N and K are compile-time (256) so all loop/epilogue addresses fold into the
// 24-bit immediate offset field of global_load/global_store (no VALU addr math).
// block = 256 threads (8 waves); wave w computes tile (blockIdx.x*16, (blockIdx.y*8+w)*16)
// A tile (16x256, shared by all 8 waves) staged in LDS once per block.
// V_WMMA_F32_16X16X4_F32 operand layouts (ISA 7.12.2):
//   A 16x4:  lanes 0-15 row M=lane {K0,K1}; lanes 16-31 row M=lane-16 {K2,K3}
//   B 4x16:  VGPR0 = rows K0 (lanes 0-15) / K2 (lanes 16-31); VGPR1 = K1 / K3
//   C 16x16: VGPR v = row v (lanes 0-15) / row v+8 (lanes 16-31), col = lane%16
// ---------------------------------------------------------------------------
__global__ void __launch_bounds__(256)
gat_gemm_wmma(const float* __restrict__ A,
              const float* __restrict__ B,
              float* __restrict__ C,
              int M) {
    __shared__ float sA[16 * APITCH];

    const int tid    = threadIdx.x;
    const int wave   = tid >> 5;
    const int lane   = tid & 31;
    const int laneLo = lane & 15;
    const int hi     = lane >> 4;               // per-lane, addressing only
    const int row0   = blockIdx.x << 4;
    const int col0   = ((blockIdx.y << 3) + wave) << 4;

    // ---- stage A tile: 16 rows x 256 cols = 4096 floats, float4 per thread x4
    {
        const float* __restrict__ abase = A + (size_t)row0 * GK;
#pragma unroll
        for (int i = 0; i < 4; ++i) {
            const int idx = tid * 4 + i * 1024;     // 0..4095, unique, 16B aligned
            const int r = idx >> 8;                 // row 0..15
            const int c = idx & 255;                // col, multiple of 4
            const float4 v = *(const float4*)(abase + r * GK + c);
            *(float4*)(&sA[r * APITCH + c]) = v;
        }
    }
    __syncthreads();

    // ---- hoisted per-lane base pointers: all loop offsets become immediates
    const int bcol = col0 + laneLo;
    const float* __restrict__ bp = B + 2 * hi * FOUT + bcol;
    const float* a0 = &sA[laneLo * APITCH + 2 * hi];

    v8f acc = {};
#pragma unroll
    for (int k = 0; k < GK; k += 4) {
        v2f a, b;
        a.x = a0[k];
        a.y = a0[k + 1];
        b.x = bp[k * FOUT];            // compile-time byte offset (<= 255*1024)
        b.y = bp[(k + 1) * FOUT];
        acc = __builtin_amdgcn_wmma_f32_16x16x4_f32(false, a, false, b,
                                                    (short)0, acc, false, false);
    }

    float* __restrict__ cp = C + (size_t)(row0 + 8 * hi) * FOUT + col0 + laneLo;
#pragma unroll
    for (int v = 0; v < 8; ++v)
        cp[v * FOUT] = acc[v];         // immediate offsets off one base
}

// ---------------------------------------------------------------------------
// a_src[n,h] = dot(h[n,h,:], att_src[h,:]);  a_dst likewise
// one thread per (node, head); att vectors are contiguous per head
// ---------------------------------------------------------------------------
__global__ void att_coef(const float* __restrict__ h,
                         const float* __restrict__ att_s,
                         const float* __restrict__ att_d,
                         float* __restrict__ as, float* __restrict__ ad,
                         int N, int heads, int C) {
    const int t = blockIdx.x * blockDim.x + threadIdx.x;
    if (t >= N * heads) return;
    const int n  = t / heads;
    const int hd = t - n * heads;
    const float* __restrict__ hp = h + (size_t)n * heads * C + (size_t)hd * C;
    const float* __restrict__ ws = att_s + (size_t)hd * C;
    const float* __restrict__ wd = att_d + (size_t)hd * C;
    float ss = 0.0f, sd = 0.0f;
#pragma unroll 8
    for (int c = 0; c < C; ++c) {
        const float v = hp[c];
        ss = fmaf(v, ws[c], ss);
        sd = fmaf(v, wd[c], sd);
    }
    as[t] = ss;
    ad[t] = sd;
}

// ---------------------------------------------------------------------------
__global__ void fillf(float* __restrict__ p, float v, int n) {
    const int t = blockIdx.x * blockDim.x + threadIdx.x;
    if (t < n) p[t] = v;
}

// ---------------------------------------------------------------------------
// decode edge (src,dst): first E entries from edge_index, then N self-loops
// ---------------------------------------------------------------------------
__device__ __forceinline__ void edge_decode(const int* __restrict__ ei, int E,
                                            int e, int& src, int& dst) {
    if (e < E) { src = ei[e]; dst = ei[E + e]; }
    else       { src = dst = e - E; }
}

// pass 1: segment max of leaky_relu(a_src[src]+a_dst[dst]) per (dst, head)
__global__ void edge_max(const int* __restrict__ ei, int E, int Etot,
                         const float* __restrict__ as, const float* __restrict__ ad,
                         float* __restrict__ m, int heads) {
    const int t = blockIdx.x * blockDim.x + threadIdx.x;
    if (t >= Etot * heads) return;
    const int e = t / heads, hd = t - e * heads;
    int src, dst; edge_decode(ei, E, e, src, dst);
    float v = as[src * heads + hd] + ad[dst * heads + hd];
    v = (v > 0.0f) ? v : 0.2f * v;
    atomicMaxF(&m[dst * heads + hd], v);
}

// pass 2: ee = exp(e - m[dst]); cache per edge; segment sum s[dst]
__global__ void edge_expsum(const int* __restrict__ ei, int E, int Etot,
                            const float* __restrict__ as, const float* __restrict__ ad,
                            const float* __restrict__ m,
                            float* __restrict__ ee, float* __restrict__ s, int heads) {
    const int t = blockIdx.x * blockDim.x + threadIdx.x;
    if (t >= Etot * heads) return;
    const int e = t / heads, hd = t - e * heads;
    int src, dst; edge_decode(ei, E, e, src, dst);
    float v = as[src * heads + hd] + ad[dst * heads + hd];
    v = (v > 0.0f) ? v : 0.2f * v;
    const float x = expf(v - m[dst * heads + hd]);
    ee[t] = x;
    atomicAdd(&s[dst * heads + hd], x);
}

// pass 3: agg[dst,f] += alpha * h[src,f]; one 256-thread block per edge,
// lane f handles feature f (coalesced row gather + coalesced row of atomics)
__global__ void edge_aggregate(const int* __restrict__ ei, int E,
                               const float* __restrict__ h,
                               const float* __restrict__ ee,
                               const float* __restrict__ s,
                               float* __restrict__ agg, int heads, int C) {
    const int e = blockIdx.x;
    const int f = threadIdx.x;           // 0..heads*C-1 == 0..255
    int src, dst; edge_decode(ei, E, e, src, dst);
    const int hd = f / C;
    const float alpha = ee[(size_t)e * heads + hd] /
                        (s[(size_t)dst * heads + hd] + 1e-16f);
    atomicAdd(&agg[(size_t)dst * FOUT + f],
              alpha * h[(size_t)src * FOUT + f]);
}

// g = elu(g + b)    (FOUT == 256, so feature index == t & 255)
__global__ void bias_elu(float* __restrict__ g, const float* __restrict__ b, int total) {
    const int t = blockIdx.x * blockDim.x + threadIdx.x;
    if (t >= total) return;
    const float v = g[t] + b[t & (FOUT - 1)];
    g[t] = (v > 0.0f) ? v : expm1f(v);
}

// out += b  (final layer: bias only, no activation)
__global__ void bias_add(float* __restrict__ out, const float* __restrict__ b, int total) {
    const int t = blockIdx.x * blockDim.x + threadIdx.x;
    if (t >= total) return;
    out[t] += b[t & (FOUT - 1)];
}

// ---------------------------------------------------------------------------
extern "C" void kernel_launch(void* const* d_in, const int* in_sizes, int n_in,
                              void* d_out, int out_size, void* d_ws, size_t ws_size,
                              hipStream_t stream) {
    const float* x    = (const float*)d_in[0];
    const int*   ei   = (const int*)d_in[1];
    const float* W1   = (const float*)d_in[2];
    const float* asw1 = (const float*)d_in[3];
    const float* adw1 = (const float*)d_in[4];
    const float* b1   = (const float*)d_in[5];
    const float* W2   = (const float*)d_in[6];
    const float* asw2 = (const float*)d_in[7];
    const float* adw2 = (const float*)d_in[8];
    const float* b2   = (const float*)d_in[9];
    float* out = (float*)d_out;

    const int N    = in_sizes[0] / FIN;   // 50000 (multiple of 16)
    const int E    = in_sizes[1] / 2;     // 800000
    const int Etot = E + N;               // + self loops

    // workspace layout (floats)
    float* h  = (float*)d_ws;                 // [N,256] GEMM output (both layers)
    float* g  = h  + (size_t)N * FOUT;        // [N,256] aggregate / elu'd layer-1 out
    float* as = g  + (size_t)N * FOUT;        // [N,4]
    float* ad = as + (size_t)N * 4;           // [N,4]
    float* m  = ad + (size_t)N * 4;           // [N,4]
    float* s  = m  + (size_t)N * 4;           // [N,4]
    float* ee = s  + (size_t)N * 4;           // [Etot,4]

    const int TB = 256;
    const float NEG_INF = -INFINITY;

    // ================= Layer 1 (heads=4, C=64) =================
    {
        dim3 gg(N / 16, FOUT / 128);
        gat_gemm_wmma<<<gg, TB, 0, stream>>>(x, W1, h, N);

        att_coef<<<(N * 4 + TB - 1) / TB, TB, 0, stream>>>(h, asw1, adw1, as, ad, N, 4, 64);

        fillf<<<(N * 4 + TB - 1) / TB, TB, 0, stream>>>(m, NEG_INF, N * 4);
        fillf<<<(N * 4 + TB - 1) / TB, TB, 0, stream>>>(s, 0.0f, N * 4);
        fillf<<<(N * FOUT + TB - 1) / TB, TB, 0, stream>>>(g, 0.0f, N * FOUT);

        edge_max   <<<(Etot * 4 + TB - 1) / TB, TB, 0, stream>>>(ei, E, Etot, as, ad, m, 4);
        edge_expsum<<<(Etot * 4 + TB - 1) / TB, TB, 0, stream>>>(ei, E, Etot, as, ad, m, ee, s, 4);
        edge_aggregate<<<Etot, TB, 0, stream>>>(ei, E, h, ee, s, g, 4, 64);

        bias_elu<<<(N * FOUT + TB - 1) / TB, TB, 0, stream>>>(g, b1, N * FOUT);
    }

    // ================= Layer 2 (heads=1, C=256) =================
    {
        dim3 gg(N / 16, FOUT / 128);
        gat_gemm_wmma<<<gg, TB, 0, stream>>>(g, W2, h, N);

        att_coef<<<(N + TB - 1) / TB, TB, 0, stream>>>(h, asw2, adw2, as, ad, N, 1, 256);

        fillf<<<(N + TB - 1) / TB, TB, 0, stream>>>(m, NEG_INF, N);
        fillf<<<(N + TB - 1) / TB, TB, 0, stream>>>(s, 0.0f, N);
        fillf<<<(N * FOUT + TB - 1) / TB, TB, 0, stream>>>(out, 0.0f, N * FOUT);

        edge_max   <<<(Etot + TB - 1) / TB, TB, 0, stream>>>(ei, E, Etot, as, ad, m, 1);
        edge_expsum<<<(Etot + TB - 1) / TB, TB, 0, stream>>>(ei, E, Etot, as, ad, m, ee, s, 1);
        edge_aggregate<<<Etot, TB, 0, stream>>>(ei, E, h, ee, s, out, 1, 256);

        bias_add<<<(N * FOUT + TB - 1) / TB, TB, 0, stream>>>(out, b2, N * FOUT);
    }
}